// GNN_1958505087509
// MI455X (gfx1250) — compile-verified
//
#include <hip/hip_runtime.h>
#include <hip/hip_bf16.h>

typedef __attribute__((ext_vector_type(2))) float v2f;
typedef __attribute__((ext_vector_type(8))) float v8f;

#define D 64

// ---------------- utility kernels ----------------

__global__ void zero_kernel(float* __restrict__ p, long long n) {
    long long t = (long long)blockIdx.x * blockDim.x + threadIdx.x;
    if (t < n) p[t] = 0.0f;
}

__global__ void deg_kernel(const int* __restrict__ dst, float* __restrict__ deg, int e) {
    int t = blockIdx.x * blockDim.x + threadIdx.x;
    if (t < e) atomicAdd(&deg[dst[t]], 1.0f);
}

__global__ void invdeg_kernel(const float* __restrict__ deg, float* __restrict__ inv, int n) {
    int t = blockIdx.x * blockDim.x + threadIdx.x;
    if (t < n) {
        float d = deg[t];
        inv[t] = (d > 0.0f) ? (1.0f / d) : 0.0f;
    }
}

// one thread per (edge, 4-feature chunk): coalesced float4 gather + 4 f32 atomics
__global__ void scatter_kernel(const float* __restrict__ x,
                               const int* __restrict__ src,
                               const int* __restrict__ dst,
                               float* __restrict__ agg, long long total) {
    long long t = (long long)blockIdx.x * blockDim.x + threadIdx.x;
    if (t >= total) return;
    int e  = (int)(t >> 4);
    int c4 = (int)(t & 15) << 2;      // feature base: 0,4,...,60
    int s = src[e];
    int d = dst[e];
    const float4 v = *(const float4*)(x + (long long)s * D + c4);
    float* o = agg + (long long)d * D + c4;
    atomicAdd(o + 0, v.x);
    atomicAdd(o + 1, v.y);
    atomicAdd(o + 2, v.z);
    atomicAdd(o + 3, v.w);
}

// ---------------- fused SAGE GEMM: out = (agg*inv_deg)@Wl + X@Wr + b (+relu) ----
// block: 128 threads = 4 waves; block handles 16 node rows; wave w -> cols 16w..16w+15
__global__ __launch_bounds__(128)
void sage_gemm_kernel(const float* __restrict__ agg,
                      const float* __restrict__ inv_deg,
                      const float* __restrict__ xin,
                      const float* __restrict__ Wl,
                      const float* __restrict__ Wr,
                      const float* __restrict__ bias,
                      float* __restrict__ out,
                      int n, int relu) {
    const int wave = threadIdx.x >> 5;         // 0..3 -> column tile
    const int lane = threadIdx.x & 31;
    const int hh   = lane >> 4;                // lane-half (0/1)
    const int l    = lane & 15;                // 0..15
    const int row0 = blockIdx.x * 16;
    const int col0 = wave * 16;

    // Stage A tiles in LDS; pad row to 65 floats so column reads hit 16 banks.
    __shared__ float sG[16][65];   // mean-aggregated neighbors
    __shared__ float sX[16][65];   // self features

    for (int i = threadIdx.x; i < 16 * D; i += 128) {
        int r = i >> 6;
        int c = i & (D - 1);
        int node = row0 + r;
        int nc = node < n ? node : (n - 1);
        sG[r][c] = agg[(long long)nc * D + c] * inv_deg[nc];
        sX[r][c] = xin[(long long)nc * D + c];
    }
    __syncthreads();

    v8f acc = {};
    #pragma unroll
    for (int kb = 0; kb < D; kb += 4) {
        const int k0 = kb + 2 * hh;            // ISA A/B layout: half selects K pair
        v2f a, b;
        // agg @ Wl
        a.x = sG[l][k0];
        a.y = sG[l][k0 + 1];
        b.x = Wl[(long long)k0 * D + col0 + l];
        b.y = Wl[(long long)(k0 + 1) * D + col0 + l];
        acc = __builtin_amdgcn_wmma_f32_16x16x4_f32(false, a, false, b,
                                                    (short)0, acc, false, false);
        // x @ Wr
        a.x = sX[l][k0];
        a.y = sX[l][k0 + 1];
        b.x = Wr[(long long)k0 * D + col0 + l];
        b.y = Wr[(long long)(k0 + 1) * D + col0 + l];
        acc = __builtin_amdgcn_wmma_f32_16x16x4_f32(false, a, false, b,
                                                    (short)0, acc, false, false);
    }

    // C/D layout: VGPR i, lane-half hh -> row (i + 8*hh), col l
    const float bb = bias[col0 + l];
    #pragma unroll
    for (int i = 0; i < 8; ++i) {
        int node = row0 + i + 8 * hh;
        if (node < n) {
            float v = acc[i] + bb;
            if (relu) v = fmaxf(v, 0.0f);
            out[(long long)node * D + col0 + l] = v;
        }
    }
}

// ---------------- launcher ----------------

extern "C" void kernel_launch(void* const* d_in, const int* in_sizes, int n_in,
                              void* d_out, int out_size, void* d_ws, size_t ws_size,
                              hipStream_t stream) {
    const float* x   = (const float*)d_in[0];
    const int*   ei  = (const int*)d_in[1];
    const float* Wl1 = (const float*)d_in[2];
    const float* Wr1 = (const float*)d_in[3];
    const float* b1  = (const float*)d_in[4];
    const float* Wl2 = (const float*)d_in[5];
    const float* Wr2 = (const float*)d_in[6];
    const float* b2  = (const float*)d_in[7];
    float* out = (float*)d_out;

    const int N = in_sizes[0] / D;
    const int E = in_sizes[1] / 2;
    const int* src = ei;          // edge_index[0]
    const int* dst = ei + E;      // edge_index[1]

    // workspace carve-up
    float* deg  = (float*)d_ws;                   // N
    float* inv  = deg + N;                        // N
    float* agg  = inv + N;                        // N*D
    float* h    = agg + (long long)N * D;         // N*D

    const int TB = 256;
    const long long ND = (long long)N * D;
    const long long SC = (long long)E * 16;

    // degrees
    zero_kernel<<<(int)((N + TB - 1) / TB), TB, 0, stream>>>(deg, N);
    deg_kernel<<<(E + TB - 1) / TB, TB, 0, stream>>>(dst, deg, E);
    invdeg_kernel<<<(N + TB - 1) / TB, TB, 0, stream>>>(deg, inv, N);

    const int gemm_grid = (N + 15) / 16;

    // layer 1: agg = scatter(x); h = relu((agg*inv)@Wl1 + x@Wr1 + b1)
    zero_kernel<<<(int)((ND + TB - 1) / TB), TB, 0, stream>>>(agg, ND);
    scatter_kernel<<<(int)((SC + TB - 1) / TB), TB, 0, stream>>>(x, src, dst, agg, SC);
    sage_gemm_kernel<<<gemm_grid, 128, 0, stream>>>(agg, inv, x, Wl1, Wr1, b1, h, N, 1);

    // layer 2: agg = scatter(h); out = (agg*inv)@Wl2 + h@Wr2 + b2
    zero_kernel<<<(int)((ND + TB - 1) / TB), TB, 0, stream>>>(agg, ND);
    scatter_kernel<<<(int)((SC + TB - 1) / TB), TB, 0, stream>>>(h, src, dst, agg, SC);
    sage_gemm_kernel<<<gemm_grid, 128, 0, stream>>>(agg, inv, h, Wl2, Wr2, b2, out, N, 0);
}